// Modular_system_34419867910535
// MI455X (gfx1250) — compile-verified
//
#include <hip/hip_runtime.h>

// MoE MLP, routed: E=24 experts, B=8192 rows, 15 -> 512 -> 512 -> 256 -> 1, f32.
// Uses CDNA5 V_WMMA_F32_16X16X4_F32 (full-precision f32 matrix core path).

#define E_   24
#define B_   8192
#define FIN_ 15
#define H1_  512
#define H2_  512
#define H3_  256
#define TM   32          // rows per workgroup tile (2 x 16-row WMMA subtiles)
#define HROW 513         // LDS row stride (512 + 1) to avoid bank conflicts

typedef float v2f __attribute__((ext_vector_type(2)));
typedef float v8f __attribute__((ext_vector_type(8)));

__device__ __forceinline__ v8f wmma4(v2f a, v2f b, v8f c) {
  // D = A(16x4 f32) * B(4x16 f32) + C(16x16 f32)
  return __builtin_amdgcn_wmma_f32_16x16x4_f32(
      /*neg_a=*/false, a, /*neg_b=*/false, b,
      /*c_mod=*/(short)0, c, /*reuse_a=*/false, /*reuse_b=*/false);
}

// ---------------- routing kernels ----------------
__global__ void moe_zero(int* counts, int* cursor) {
  int t = threadIdx.x;
  if (t < E_) { counts[t] = 0; cursor[t] = 0; }
}

__global__ void moe_hist(const int* __restrict__ idx, int* counts) {
  int b = blockIdx.x * blockDim.x + threadIdx.x;
  if (b < B_) atomicAdd(&counts[idx[b]], 1);
}

__global__ void moe_scan(const int* __restrict__ counts, int* offsets, int* cursor) {
  if (threadIdx.x == 0) {
    int acc = 0;
    for (int e = 0; e < E_; ++e) { offsets[e] = acc; cursor[e] = acc; acc += counts[e]; }
    offsets[E_] = acc;
  }
}

__global__ void moe_scatter(const int* __restrict__ idx, int* cursor, int* perm) {
  int b = blockIdx.x * blockDim.x + threadIdx.x;
  if (b < B_) { int p = atomicAdd(&cursor[idx[b]], 1); perm[p] = b; }
}

// ---------------- fused routed MLP ----------------
__global__ __launch_bounds__(256)
void moe_mlp(const float* __restrict__ feats,
             const float* __restrict__ W1, const float* __restrict__ b1,
             const float* __restrict__ W2, const float* __restrict__ b2,
             const float* __restrict__ W3, const float* __restrict__ b3,
             const float* __restrict__ W4, const float* __restrict__ b4,
             const int* __restrict__ offsets, const int* __restrict__ perm,
             float* __restrict__ out) {
  const int e     = blockIdx.y;
  const int start = offsets[e];
  const int cnt   = offsets[e + 1] - start;
  const int t0    = blockIdx.x * TM;
  if (t0 >= cnt) return;                       // uniform early exit
  const int nrows = min(TM, cnt - t0);

  __shared__ int   rows[TM];
  __shared__ float Xs[TM][16];                 // input tile, K padded 15 -> 16
  __shared__ float Ha[TM][HROW];               // H1, later reused for H3
  __shared__ float Hb[TM][HROW];               // H2
  __shared__ float red[TM][8];

  const int tid = threadIdx.x;
  if (tid < TM) rows[tid] = (tid < nrows) ? perm[start + t0 + tid] : -1;
  __syncthreads();

  for (int i2 = tid; i2 < TM * 16; i2 += 256) {
    int r = i2 >> 4, c = i2 & 15;
    float v = 0.f;
    int rr = rows[r];
    if (rr >= 0 && c < FIN_) v = feats[rr * FIN_ + c];
    Xs[r][c] = v;
  }
  __syncthreads();

  const int wave = tid >> 5;                   // 8 wave32s per block
  const int lane = tid & 31;
  const int lm   = lane & 15;                  // M (A) / N (B,C) within tile
  const int hi   = lane >> 4;                  // lane half
  const int lk   = hi * 2;                     // K base for this lane (f32 16x16x4 layout)

  // ---- Layer 1: [TM x 16] x W1[15 x 512] -> H1 ----
  {
    const float* W1e = W1 + e * FIN_ * H1_;
    for (int i = 0; i < 4; ++i) {              // 8 waves x 4 tiles = 512 cols
      const int n = (wave * 4 + i) * 16 + lm;
      v8f acc0 = {}; v8f acc1 = {};
      for (int kk = 0; kk < 4; ++kk) {
        const int k = kk * 4 + lk;
        v2f a0, a1, bf;
        a0.x = Xs[lm][k];        a0.y = Xs[lm][k + 1];
        a1.x = Xs[16 + lm][k];   a1.y = Xs[16 + lm][k + 1];
        bf.x = (k     < FIN_) ? W1e[k * H1_ + n]       : 0.f;
        bf.y = (k + 1 < FIN_) ? W1e[(k + 1) * H1_ + n] : 0.f;
        acc0 = wmma4(a0, bf, acc0);
        acc1 = wmma4(a1, bf, acc1);
      }
      const float bias = b1[e * H1_ + n];
#pragma unroll
      for (int v = 0; v < 8; ++v) {            // C layout: VGPR v -> M = v + hi*8
        const int m = v + hi * 8;
        float x0 = acc0[v] + bias; Ha[m][n]      = x0 > 0.f ? x0 : 0.f;
        float x1 = acc1[v] + bias; Ha[16 + m][n] = x1 > 0.f ? x1 : 0.f;
      }
    }
  }
  __syncthreads();

  // ---- Layer 2: H1[TM x 512] x W2[512 x 512] -> H2 ----
  {
    const float* W2e = W2 + e * H1_ * H2_;
    for (int i = 0; i < 4; ++i) {
      const int n = (wave * 4 + i) * 16 + lm;
      v8f acc0 = {}; v8f acc1 = {};
#pragma unroll 4
      for (int k0 = 0; k0 < H1_; k0 += 4) {
        const int k = k0 + lk;
        v2f a0, a1, bf;
        a0.x = Ha[lm][k];        a0.y = Ha[lm][k + 1];
        a1.x = Ha[16 + lm][k];   a1.y = Ha[16 + lm][k + 1];
        bf.x = W2e[k * H2_ + n];
        bf.y = W2e[(k + 1) * H2_ + n];
        acc0 = wmma4(a0, bf, acc0);
        acc1 = wmma4(a1, bf, acc1);
      }
      const float bias = b2[e * H2_ + n];
#pragma unroll
      for (int v = 0; v < 8; ++v) {
        const int m = v + hi * 8;
        float x0 = acc0[v] + bias; Hb[m][n]      = x0 > 0.f ? x0 : 0.f;
        float x1 = acc1[v] + bias; Hb[16 + m][n] = x1 > 0.f ? x1 : 0.f;
      }
    }
  }
  __syncthreads();

  // ---- Layer 3: H2[TM x 512] x W3[512 x 256] -> H3 (into Ha) ----
  {
    const float* W3e = W3 + e * H2_ * H3_;
    for (int i = 0; i < 2; ++i) {              // 8 waves x 2 tiles = 256 cols
      const int n = (wave * 2 + i) * 16 + lm;
      v8f acc0 = {}; v8f acc1 = {};
#pragma unroll 4
      for (int k0 = 0; k0 < H2_; k0 += 4) {
        const int k = k0 + lk;
        v2f a0, a1, bf;
        a0.x = Hb[lm][k];        a0.y = Hb[lm][k + 1];
        a1.x = Hb[16 + lm][k];   a1.y = Hb[16 + lm][k + 1];
        bf.x = W3e[k * H3_ + n];
        bf.y = W3e[(k + 1) * H3_ + n];
        acc0 = wmma4(a0, bf, acc0);
        acc1 = wmma4(a1, bf, acc1);
      }
      const float bias = b3[e * H3_ + n];
#pragma unroll
      for (int v = 0; v < 8; ++v) {
        const int m = v + hi * 8;
        float x0 = acc0[v] + bias; Ha[m][n]      = x0 > 0.f ? x0 : 0.f;
        float x1 = acc1[v] + bias; Ha[16 + m][n] = x1 > 0.f ? x1 : 0.f;
      }
    }
  }
  __syncthreads();

  // ---- Layer 4: H3[TM x 256] x W4[256 x 1] + b4 -> out ----
  {
    const float* W4e = W4 + e * H3_;
    const int r   = tid >> 3;                  // 0..31
    const int seg = tid & 7;                   // 8 segments x 32 cols
    const int c0  = seg * 32;
    float s = 0.f;
#pragma unroll 8
    for (int c = 0; c < 32; ++c) s += Ha[r][c0 + c] * W4e[c0 + c];
    red[r][seg] = s;
  }
  __syncthreads();
  if (tid < TM) {
    float s = b4[e];
#pragma unroll
    for (int j = 0; j < 8; ++j) s += red[tid][j];
    if (tid < nrows) out[rows[tid]] = s;       // out is [B,1]
  }
}

extern "C" void kernel_launch(void* const* d_in, const int* in_sizes, int n_in,
                              void* d_out, int out_size, void* d_ws, size_t ws_size,
                              hipStream_t stream) {
  (void)in_sizes; (void)n_in; (void)out_size; (void)ws_size;
  const float* feats = (const float*)d_in[0];
  const int*   midx  = (const int*)  d_in[1];
  const float* W1 = (const float*)d_in[2]; const float* b1 = (const float*)d_in[3];
  const float* W2 = (const float*)d_in[4]; const float* b2 = (const float*)d_in[5];
  const float* W3 = (const float*)d_in[6]; const float* b3 = (const float*)d_in[7];
  const float* W4 = (const float*)d_in[8]; const float* b4 = (const float*)d_in[9];
  float* out = (float*)d_out;

  int* ws      = (int*)d_ws;
  int* counts  = ws;
  int* cursor  = ws + E_;
  int* offsets = ws + 2 * E_;
  int* perm    = ws + 2 * E_ + (E_ + 1);

  moe_zero<<<1, 32, 0, stream>>>(counts, cursor);
  moe_hist<<<(B_ + 255) / 256, 256, 0, stream>>>(midx, counts);
  moe_scan<<<1, 32, 0, stream>>>(counts, offsets, cursor);
  moe_scatter<<<(B_ + 255) / 256, 256, 0, stream>>>(midx, cursor, perm);

  dim3 grid((B_ + TM - 1) / TM, E_);
  moe_mlp<<<grid, 256, 0, stream>>>(feats, W1, b1, W2, b2, W3, b3, W4, b4,
                                    offsets, perm, out);
}